// GATv2Model_4337916969224
// MI455X (gfx1250) — compile-verified
//
#include <hip/hip_runtime.h>
#include <hip/hip_bf16.h>

typedef __attribute__((ext_vector_type(2))) float v2f;
typedef __attribute__((ext_vector_type(8))) float v8f;

#define NN 207
#define DD 128
#define BB 16
#define GF_LN   1
#define GF_RELU 2
#define GF_RES  4

__device__ __forceinline__ v8f wmma4(v2f a, v2f b, v8f c) {
  // D = A(16x4 f32) * B(4x16 f32) + C(16x16 f32)
  return __builtin_amdgcn_wmma_f32_16x16x4_f32(false, a, false, b, (short)0, c,
                                               false, false);
}

// async global->LDS 16B copy, GVS addressing: mem = saddr + voff; lds = lds_off
__device__ __forceinline__ void async_cp16(uint32_t lds_off, uint32_t voff,
                                           const void* saddr) {
  asm volatile("global_load_async_to_lds_b128 %0, %1, %2"
               :: "v"(lds_off), "v"(voff), "s"(saddr) : "memory");
}
__device__ __forceinline__ void wait_async_le4() {
  asm volatile("s_wait_asynccnt 0x4" ::: "memory");
}
__device__ __forceinline__ void wait_async_0() {
  asm volatile("s_wait_asynccnt 0x0" ::: "memory");
}
__device__ __forceinline__ uint32_t lds_lo32(const void* p) {
  return (uint32_t)(uintptr_t)p;  // addr[31:0] of a shared ptr = LDS byte offset
}

// ---------------- input MLP layer 1: y = relu(LN(x @ W1^T + b1)) -----------
__global__ void in1_kernel(const float* __restrict__ x, const float* __restrict__ W1,
                           const float* __restrict__ b1, const float* __restrict__ g,
                           const float* __restrict__ bt, float* __restrict__ out) {
  const int row = blockIdx.x;   // 0..3311
  const int n = threadIdx.x;    // 0..127
  const float* xr = x + row * 6;
  float acc = b1[n];
#pragma unroll
  for (int k = 0; k < 6; ++k) acc += xr[k] * W1[n * 6 + k];
  __shared__ float sv[128];
  __shared__ float r1[32], r2[32];
  __shared__ float ms[2];
  sv[n] = acc;
  __syncthreads();
  if (n < 32) {
    float s = 0.f, s2 = 0.f;
#pragma unroll
    for (int q = 0; q < 4; ++q) { float v = sv[n * 4 + q]; s += v; s2 += v * v; }
    r1[n] = s; r2[n] = s2;
  }
  __syncthreads();
  if (n == 0) {
    float s = 0.f, s2 = 0.f;
    for (int q = 0; q < 32; ++q) { s += r1[q]; s2 += r2[q]; }
    float mean = s * (1.f / 128.f);
    float var = s2 * (1.f / 128.f) - mean * mean;
    ms[0] = mean; ms[1] = rsqrtf(var + 1e-5f);
  }
  __syncthreads();
  float v = (acc - ms[0]) * ms[1] * g[n] + bt[n];
  out[row * 128 + n] = fmaxf(v, 0.f);
}

// ------- WMMA GEMM: Y[M,128] = X[M,128] @ W[128,128]^T + bias (+LN/ReLU/res)
// X tile (16x128) staged once per block via async global->LDS (8x reuse by waves)
__global__ void gemm128(const float* __restrict__ X, const float* __restrict__ W,
                        const float* __restrict__ bias, const float* __restrict__ gamma,
                        const float* __restrict__ beta, const float* __restrict__ res,
                        float* __restrict__ Y, int flags) {
  const int tid = threadIdx.x;
  const int wv = tid >> 5;       // wave = N-tile (8 waves, 128 cols)
  const int lane = tid & 31;
  const int l16 = lane & 15;
  const int half = lane >> 4;
  const int m0 = blockIdx.x * 16;
  const int n0 = wv * 16;

  __shared__ float sx[16][132];  // padded: bank-conflict-free A-fragment reads
  {
    const uint32_t ldsb = lds_lo32(&sx[0][0]);
#pragma unroll
    for (int qq = 0; qq < 2; ++qq) {
      int q = tid + 256 * qq;          // 0..511 16B-chunks
      int r = q >> 5;                  // row 0..15
      int c4 = (q & 31) << 2;          // col 0,4,...,124
      uint32_t voff = (uint32_t)(((m0 + r) * 128 + c4) * 4);
      uint32_t la = ldsb + (uint32_t)((r * 132 + c4) * 4);
      async_cp16(la, voff, X);
    }
  }
  wait_async_0();
  __syncthreads();

  const float* wp = W + (n0 + l16) * 128 + half * 2;  // B: N=l16, K pair (W[n,k])
  const float* ap = &sx[l16][half * 2];
  v8f c = {0.f, 0.f, 0.f, 0.f, 0.f, 0.f, 0.f, 0.f};
#pragma unroll 8
  for (int k = 0; k < 128; k += 4) {
    v2f a, b;
    a.x = ap[k]; a.y = ap[k + 1];
    b.x = wp[k]; b.y = wp[k + 1];
    c = wmma4(a, b, c);
  }
  const float bv = bias[n0 + l16];
#pragma unroll
  for (int r = 0; r < 8; ++r) c[r] += bv;

  if (!(flags & GF_LN)) {  // plain store (C layout: VGPR r -> M=r+half*8, N=l16)
#pragma unroll
    for (int r = 0; r < 8; ++r) {
      float v = c[r];
      if (flags & GF_RELU) v = fmaxf(v, 0.f);
      int m = m0 + r + half * 8;
      if (flags & GF_RES) v += res[m * 128 + n0 + l16];
      Y[m * 128 + n0 + l16] = v;
    }
    return;
  }
  __shared__ float tile[16][128];
  __shared__ float ps[16][17], ps2[16][17];
  __shared__ float mean[16], rstd[16];
#pragma unroll
  for (int r = 0; r < 8; ++r) tile[r + half * 8][n0 + l16] = c[r];
  __syncthreads();
  const int row = tid & 15, part = tid >> 4;
  float s = 0.f, s2 = 0.f;
#pragma unroll
  for (int q = 0; q < 8; ++q) { float v = tile[row][part * 8 + q]; s += v; s2 += v * v; }
  ps[row][part] = s; ps2[row][part] = s2;
  __syncthreads();
  if (part == 0) {
    float t = 0.f, t2 = 0.f;
    for (int q = 0; q < 16; ++q) { t += ps[row][q]; t2 += ps2[row][q]; }
    float mn = t * (1.f / 128.f);
    mean[row] = mn;
    rstd[row] = rsqrtf(t2 * (1.f / 128.f) - mn * mn + 1e-5f);
  }
  __syncthreads();
#pragma unroll
  for (int q = 0; q < 8; ++q) {
    int col = part * 8 + q;
    float v = (tile[row][col] - mean[row]) * rstd[row] * gamma[col] + beta[col];
    if (flags & GF_RELU) v = fmaxf(v, 0.f);
    int m = m0 + row;
    if (flags & GF_RES) v += res[m * 128 + col];
    Y[m * 128 + col] = v;
  }
}

// --------- edge-category transform: ecat[l*12+e, n] = emb[e] . We[l][n] -----
__global__ void edge_kernel(const float* __restrict__ emb, const float* __restrict__ We,
                            float* __restrict__ ecat) {
  const int le = blockIdx.x;       // 0..47 (layer*12+cat)
  const int l = le / 12, e = le % 12;
  const int n = threadIdx.x;
  const float* ep = emb + e * 128;
  const float* wp = We + (l * 128 + n) * 128;
  float acc = 0.f;
#pragma unroll 8
  for (int k = 0; k < 128; ++k) acc += ep[k] * wp[k];
  ecat[le * 128 + n] = acc;
}

// issue one 32-row stage of gs (32x128 f32 = 16KB) as 4 async b128 per thread
__device__ __forceinline__ void stage_gs(const float* gsbase, int sb,
                                         uint32_t ldsb, int tid) {
#pragma unroll
  for (int qq = 0; qq < 4; ++qq) {
    int q = tid + 256 * qq;          // 0..1023 16B-chunks
    int r = q >> 5;                  // row-in-stage 0..31
    int c4 = (q & 31) << 2;          // col 0..124
    int ir = sb + r;
    ir = (ir < NN) ? ir : (NN - 1);  // clamp tail rows (finite dummy data)
    uint32_t voff = (uint32_t)((ir * 128 + c4) * 4);
    uint32_t la = ldsb + (uint32_t)(q << 4);
    async_cp16(la, voff, gsbase);
  }
}

// ------- fused GATv2 attention: scores + online softmax + WMMA aggregation --
// grid (13, B); block 256 = 8 waves (wave = head); tile = 16 targets j.
// gs rows double-buffered in LDS via async copies (ASYNCcnt pipelined).
__global__ void attn_kernel(const float* __restrict__ gs, const float* __restrict__ gd,
                            const int* __restrict__ cat, const float* __restrict__ ee,
                            const float* __restrict__ att, const float* __restrict__ cb,
                            float* __restrict__ o) {
  const int b = blockIdx.y;
  const int j0 = blockIdx.x * 16;
  const int tid = threadIdx.x;
  const int h = tid >> 5;
  const int lane = tid & 31;
  const int jj = lane & 15;
  const int half = lane >> 4;

  __shared__ float s_ee[12 * 128];      // 12 edge categories transformed (6 KB)
  __shared__ float sgs[2][32 * 128];    // double-buffered gs stage (32 KB)
  for (int idx = tid; idx < 12 * 128; idx += 256) s_ee[idx] = ee[idx];

  const float* gsbase = gs + (size_t)b * NN * 128;
  const int NSTAGE = (NN + 31) / 32;    // 7
  stage_gs(gsbase, 0, lds_lo32(&sgs[0][0]), tid);   // prologue

  const int jg = j0 + jj;
  const int j = (jg < NN) ? jg : (NN - 1);  // clamp; invalid columns never stored
  float gdr[16], attr[16];
  const float* gdp = gd + (b * NN + j) * 128 + h * 16;
#pragma unroll
  for (int cc = 0; cc < 16; ++cc) { gdr[cc] = gdp[cc]; attr[cc] = att[h * 16 + cc]; }

  float m = -3.0e38f, d = 0.f;
  v8f acc = {0.f, 0.f, 0.f, 0.f, 0.f, 0.f, 0.f, 0.f};

  for (int st = 0; st < NSTAGE; ++st) {
    if (st + 1 < NSTAGE) {
      stage_gs(gsbase, (st + 1) * 32, lds_lo32(&sgs[(st + 1) & 1][0]), tid);
      wait_async_le4();                 // stage st complete, st+1 in flight
    } else {
      wait_async_0();
    }
    __syncthreads();
    const float* gl = &sgs[st & 1][0];
    const int base = st * 32;

    for (int icl = 0; icl < 32; icl += 4) {
      const int ic = base + icl;
      if (ic >= NN) break;              // uniform across block
      const int ia = ic + half * 2;     // lane<16 -> K=0,1 ; lane>=16 -> K=2,3
      const int ib = ia + 1;
      const int ra = icl + half * 2;    // row-in-stage
      float s0 = -3.0e38f, s1 = -3.0e38f;
      if (ia < NN) {
        const int ct = cat[ia * NN + j];
        const float* gsp = gl + ra * 128 + h * 16;
        const float* eep = s_ee + ct * 128 + h * 16;
        float a0 = 0.f;
#pragma unroll
        for (int cc = 0; cc < 16; ++cc) {
          float v = gsp[cc] + gdr[cc] + eep[cc];
          v = (v > 0.f) ? v : 0.2f * v;          // leaky_relu 0.2
          a0 += attr[cc] * v;
        }
        s0 = a0;
      }
      if (ib < NN) {
        const int ct = cat[ib * NN + j];
        const float* gsp = gl + (ra + 1) * 128 + h * 16;
        const float* eep = s_ee + ct * 128 + h * 16;
        float a1 = 0.f;
#pragma unroll
        for (int cc = 0; cc < 16; ++cc) {
          float v = gsp[cc] + gdr[cc] + eep[cc];
          v = (v > 0.f) ? v : 0.2f * v;
          a1 += attr[cc] * v;
        }
        s1 = a1;
      }
      // online softmax update (per target column jj, replicated in both halves)
      float cm = fmaxf(s0, s1);
      cm = fmaxf(cm, __shfl_xor(cm, 16));
      const float mn = fmaxf(m, cm);
      const float scale = __expf(m - mn);
      const float e0 = __expf(s0 - mn);
      const float e1 = __expf(s1 - mn);
      float es = e0 + e1;
      es += __shfl_xor(es, 16);
      d = d * scale + es;
      m = mn;
      // rescale accumulator rows: C-layout row of reg r is jj = r + half*8
#pragma unroll
      for (int r = 0; r < 8; ++r) acc[r] *= __shfl(scale, r + half * 8);
      // A frag: exp-scores already in layout; B frag from LDS stage
      v2f af, bf;
      af.x = e0; af.y = e1;
      bf.x = gl[ra * 128 + h * 16 + jj];
      bf.y = gl[(ra + 1) * 128 + h * 16 + jj];
      acc = wmma4(af, bf, acc);
    }
    __syncthreads();
  }
  const float cbr = cb[h * 16 + jj];
#pragma unroll
  for (int r = 0; r < 8; ++r) {
    const int jr = r + half * 8;
    const float dr = __shfl(d, jr);
    const int jglob = j0 + jr;
    if (jglob < NN)
      o[(b * NN + jglob) * 128 + h * 16 + jj] = acc[r] / dr + cbr;
  }
}

// --------------------- output head: [3312,128] @ [2,128]^T + ob -------------
__global__ void out_kernel(const float* __restrict__ hh, const float* __restrict__ oW,
                           const float* __restrict__ ob, float* __restrict__ out) {
  const int idx = blockIdx.x * 256 + threadIdx.x;
  if (idx >= BB * NN * 2) return;
  const int row = idx >> 1, oo = idx & 1;
  const float* hp = hh + row * 128;
  const float* wp = oW + oo * 128;
  float acc = ob[oo];
#pragma unroll 8
  for (int k = 0; k < 128; ++k) acc += hp[k] * wp[k];
  out[idx] = acc;
}

extern "C" void kernel_launch(void* const* d_in, const int* in_sizes, int n_in,
                              void* d_out, int out_size, void* d_ws, size_t ws_size,
                              hipStream_t stream) {
  const float* x     = (const float*)d_in[0];
  const int*   cat   = (const int*)  d_in[1];
  const float* emb   = (const float*)d_in[2];
  const float* in1_W = (const float*)d_in[3];
  const float* in1_b = (const float*)d_in[4];
  const float* ln1_g = (const float*)d_in[5];
  const float* ln1_b = (const float*)d_in[6];
  const float* in2_W = (const float*)d_in[7];
  const float* in2_b = (const float*)d_in[8];
  const float* ln2_g = (const float*)d_in[9];
  const float* ln2_b = (const float*)d_in[10];
  const float* Wl    = (const float*)d_in[11];
  const float* bl    = (const float*)d_in[12];
  const float* Wr    = (const float*)d_in[13];
  const float* br    = (const float*)d_in[14];
  const float* We    = (const float*)d_in[15];
  const float* att   = (const float*)d_in[16];
  const float* cb    = (const float*)d_in[17];
  const float* pW    = (const float*)d_in[18];
  const float* pb    = (const float*)d_in[19];
  const float* lng   = (const float*)d_in[20];
  const float* lnb   = (const float*)d_in[21];
  const float* oW    = (const float*)d_in[22];
  const float* ob    = (const float*)d_in[23];
  float* out = (float*)d_out;
  float* ws  = (float*)d_ws;

  const int R = BB * NN;  // 3312 rows, divisible by 16
  float* hA   = ws;                 // current h
  float* hB   = hA + R * DD;        // ping-pong partner (also holds MLP stage 1)
  float* gsb  = hB + R * DD;
  float* gdb  = gsb + R * DD;
  float* obuf = gdb + R * DD;
  float* ecat = obuf + R * DD;      // [4][12][128]

  in1_kernel<<<R, 128, 0, stream>>>(x, in1_W, in1_b, ln1_g, ln1_b, hB);
  gemm128<<<R / 16, 256, 0, stream>>>(hB, in2_W, in2_b, ln2_g, ln2_b, nullptr, hA, GF_LN);
  edge_kernel<<<48, 128, 0, stream>>>(emb, We, ecat);

  float* hc = hA; float* hn = hB;
  for (int l = 0; l < 4; ++l) {
    gemm128<<<R / 16, 256, 0, stream>>>(hc, Wl + l * DD * DD, bl + l * DD,
                                        nullptr, nullptr, nullptr, gsb, 0);
    gemm128<<<R / 16, 256, 0, stream>>>(hc, Wr + l * DD * DD, br + l * DD,
                                        nullptr, nullptr, nullptr, gdb, 0);
    attn_kernel<<<dim3(13, BB), 256, 0, stream>>>(gsb, gdb, cat, ecat + l * 12 * DD,
                                                  att + l * DD, cb + l * DD, obuf);
    gemm128<<<R / 16, 256, 0, stream>>>(obuf, pW + l * DD * DD, pb + l * DD,
                                        lng + l * DD, lnb + l * DD, hc, hn,
                                        GF_LN | GF_RELU | GF_RES);
    float* t = hc; hc = hn; hn = t;
  }
  out_kernel<<<(R * 2 + 255) / 256, 256, 0, stream>>>(hc, oW, ob, out);
}